// ROIAlign_29755533426740
// MI455X (gfx1250) — compile-verified
//
#include <hip/hip_runtime.h>
#include <stdint.h>

// ROIAlign (FPN, tf.crop_and_resize semantics) for MI455X / gfx1250.
// Memory-bound gather: no matmul -> WMMA not applicable. CDNA5 path used:
// async global->LDS DMA (global_load_async_to_lds_b128 + ASYNCcnt) with
// per-wave double buffering, plus non-temporal output stores so the 50 MB
// write stream does not evict the ~150 MB feature pyramid from the 192 MB L2.

#define ROI_OUT 7
#define NPOS    49           // 7x7
#define NCH     256          // channels
#define HALF    128          // channels per task (32 lanes x float4)
#define NTASK   98           // 49 positions x 2 channel halves
#define NWAVE   8            // 256 threads / wave32

typedef float v4f __attribute__((ext_vector_type(4)));   // native clang vector

__device__ __forceinline__ unsigned lds_off(const void* p) {
    // On gfx1250 the LDS aperture maps generic addr[31:0] directly to the
    // LDS byte offset, so truncation yields the ds-space address.
    return (unsigned)(uintptr_t)p;
}

__global__ __launch_bounds__(256) void roialign_fused(
    const float* __restrict__ p2, const float* __restrict__ p3,
    const float* __restrict__ p4, const float* __restrict__ p5,
    const float* __restrict__ rois, const int* __restrict__ bidx,
    float* __restrict__ out, int N)
{
    // [wave][double-buffer][corner tl,tr,bl,br][HALF channels] = 32 KB
    __shared__ float stage[NWAVE][2][4][HALF];
    __shared__ int   sy0[ROI_OUT], sy1[ROI_OUT], sx0[ROI_OUT], sx1[ROI_OUT];
    __shared__ float sly[ROI_OUT], slx[ROI_OUT];
    __shared__ int   svy[ROI_OUT], svx[ROI_OUT];

    const int n    = blockIdx.x;
    const int tid  = threadIdx.x;
    const int wave = tid >> 5;
    const int lane = tid & 31;

    // ---- per-ROI setup (all threads; uniform -> compiler scalarizes) ----
    const float rx1 = rois[4 * n + 0], ry1 = rois[4 * n + 1];
    const float rx2 = rois[4 * n + 2], ry2 = rois[4 * n + 3];
    const float rw = rx2 - rx1, rh = ry2 - ry1;
    // levels = round(log(sqrt(h*w)/224)/log(2) + 4), clipped to [2,5]
    float lvl = logf(sqrtf(rh * rw) / 224.0f) / 0.69314718f + 4.0f;
    int li = (int)rintf(lvl);
    li = (li < 2) ? 2 : (li > 5 ? 5 : li);
    li -= 2;                                   // 0..3
    const int    dim  = 256 >> li;             // H == W per level
    const float  strd = (float)(4 << li);      // stride per level
    const float* fm   = (li == 0) ? p2 : (li == 1) ? p3 : (li == 2) ? p4 : p5;
    const int    b    = bidx[n];
    const float* fmb  = fm + (size_t)b * (size_t)dim * (size_t)dim * NCH;

    if (tid < ROI_OUT) {                       // y axis tables
        const float fH = (float)dim;
        const float b1 = (ry1 / strd) / fH;
        const float b2 = (ry2 / strd) / fH;
        const float g  = (float)tid / 6.0f;
        const float iy = (b1 + (b2 - b1) * g) * (fH - 1.0f);
        const float fl = floorf(iy);
        sly[tid] = iy - fl;
        svy[tid] = (iy >= 0.0f) && (iy <= fH - 1.0f);
        sy0[tid] = (int)fminf(fmaxf(fl, 0.0f), fH - 1.0f);
        sy1[tid] = (int)fminf(fmaxf(ceilf(iy), 0.0f), fH - 1.0f);
    } else if (tid >= 32 && tid < 32 + ROI_OUT) {   // x axis tables
        const int   a  = tid - 32;
        const float fW = (float)dim;
        const float b1 = (rx1 / strd) / fW;
        const float b2 = (rx2 / strd) / fW;
        const float g  = (float)a / 6.0f;
        const float ix = (b1 + (b2 - b1) * g) * (fW - 1.0f);
        const float fl = floorf(ix);
        slx[a] = ix - fl;
        svx[a] = (ix >= 0.0f) && (ix <= fW - 1.0f);
        sx0[a] = (int)fminf(fmaxf(fl, 0.0f), fW - 1.0f);
        sx1[a] = (int)fminf(fmaxf(ceilf(ix), 0.0f), fW - 1.0f);
    }
    if (tid == 0) {
        // second tuple output: batch_indices bits appended after the floats
        ((int*)out)[(size_t)N * NPOS * NCH + n] = b;
    }
    __syncthreads();

    const unsigned my_lds = lds_off(&stage[wave][0][0][0]);

    // issue: async-DMA the 4 corner float4s of one (position, half) into LDS.
    // Default (RT) temporal policy: corner rows are reused across grid cells
    // and ROIs; the pyramid nearly fits MI455X's 192 MB L2.
    auto issue = [&](int task, int sel) {
        const int p  = task >> 1, half = task & 1;
        const int gy = p / 7,     gx   = p - gy * 7;
        const int y0 = sy0[gy], y1 = sy1[gy];
        const int x0 = sx0[gx], x1 = sx1[gx];
        const int cb = half * HALF + lane * 4;
        const float* r0 = fmb + ((size_t)(y0 * dim + x0)) * NCH + cb; // tl
        const float* r1 = fmb + ((size_t)(y0 * dim + x1)) * NCH + cb; // tr
        const float* r2 = fmb + ((size_t)(y1 * dim + x0)) * NCH + cb; // bl
        const float* r3 = fmb + ((size_t)(y1 * dim + x1)) * NCH + cb; // br
        const unsigned l0 = my_lds + (unsigned)((sel * 4 * HALF + lane * 4) * 4);
        const unsigned l1 = l0 + (unsigned)(HALF * 4);
        const unsigned l2 = l0 + (unsigned)(2 * HALF * 4);
        const unsigned l3 = l0 + (unsigned)(3 * HALF * 4);
        asm volatile("global_load_async_to_lds_b128 %0, %1, off"
                     :: "v"(l0), "v"(r0) : "memory");
        asm volatile("global_load_async_to_lds_b128 %0, %1, off"
                     :: "v"(l1), "v"(r1) : "memory");
        asm volatile("global_load_async_to_lds_b128 %0, %1, off"
                     :: "v"(l2), "v"(r2) : "memory");
        asm volatile("global_load_async_to_lds_b128 %0, %1, off"
                     :: "v"(l3), "v"(r3) : "memory");
    };

    // consume: bilinear combine (exact reference lerp order) + masked store.
    // Output is write-once streaming -> non-temporal store (th:TH_STORE_NT)
    // to avoid evicting the feature pyramid from L2.
    auto consume = [&](int task, int sel) {
        const int p  = task >> 1, half = task & 1;
        const int gy = p / 7,     gx   = p - gy * 7;
        const float ly = sly[gy], lx = slx[gx];
        const bool  valid = svy[gy] && svx[gx];
        const v4f tl = *(const v4f*)&stage[wave][sel][0][lane * 4];
        const v4f tr = *(const v4f*)&stage[wave][sel][1][lane * 4];
        const v4f bl = *(const v4f*)&stage[wave][sel][2][lane * 4];
        const v4f br = *(const v4f*)&stage[wave][sel][3][lane * 4];
        v4f o;
        {
            float top, bot;
            top = tl.x + (tr.x - tl.x) * lx; bot = bl.x + (br.x - bl.x) * lx;
            o.x = top + (bot - top) * ly;
            top = tl.y + (tr.y - tl.y) * lx; bot = bl.y + (br.y - bl.y) * lx;
            o.y = top + (bot - top) * ly;
            top = tl.z + (tr.z - tl.z) * lx; bot = bl.z + (br.z - bl.z) * lx;
            o.z = top + (bot - top) * ly;
            top = tl.w + (tr.w - tl.w) * lx; bot = bl.w + (br.w - bl.w) * lx;
            o.w = top + (bot - top) * ly;
        }
        if (!valid) { o = (v4f)(0.0f); }
        v4f* dst = (v4f*)(out + (((size_t)n * NPOS + p) * NCH
                                 + half * HALF + lane * 4));
        __builtin_nontemporal_store(o, dst);
    };

    // ---- software pipeline, double-buffered in LDS, ASYNCcnt-synced ----
    int t = wave, sel = 0;
    if (t < NTASK) issue(t, 0);
    while (t < NTASK) {
        const int tn = t + NWAVE;
        if (tn < NTASK) {
            issue(tn, sel ^ 1);
            // 4 in-flight ops belong to the *next* stage; oldest 4 (== this
            // stage) are guaranteed complete once ASYNCcnt <= 4 (in-order).
            asm volatile("s_wait_asynccnt 4" ::: "memory");
        } else {
            asm volatile("s_wait_asynccnt 0" ::: "memory");
        }
        consume(t, sel);
        t = tn;
        sel ^= 1;
    }
}

extern "C" void kernel_launch(void* const* d_in, const int* in_sizes, int n_in,
                              void* d_out, int out_size, void* d_ws, size_t ws_size,
                              hipStream_t stream) {
    const float* p2   = (const float*)d_in[0];
    const float* p3   = (const float*)d_in[1];
    const float* p4   = (const float*)d_in[2];
    const float* p5   = (const float*)d_in[3];
    const float* rois = (const float*)d_in[4];
    const int*   bi   = (const int*)d_in[5];
    const int N = in_sizes[4] / 4;     // rois is [N,4]

    roialign_fused<<<dim3(N), dim3(256), 0, stream>>>(
        p2, p3, p4, p5, rois, bi, (float*)d_out, N);
}